// IdealLearner_90177133347004
// MI455X (gfx1250) — compile-verified
//
#include <hip/hip_runtime.h>
#include <cstdint>
#include <cstddef>

// ---------------------------------------------------------------------------
// IdealLearner on MI455X (gfx1250, wave32)
//
//  Phase 1 : blocked sequential recurrence. Per 16-step block:
//            - probe chain on 64 threads (no barriers, ~5 ops + exp on chain)
//            - state update R <- mem^16 R + W@TMg as 64x16 @ 16x256 f32 WMMA
//            Emits c_t[a] and per-block carries R_b[a,n].
//  Phase 2 : scan-as-lower-triangular-GEMM on V_WMMA_F32_16X16X4_F32,
//            bandwidth-bound 268 MB output stream.
// ---------------------------------------------------------------------------

#define NALGOS 64
#define NTASKS 256
#define SEQLEN 4096
#define TP1    4097        // SEQLEN + 1 (output time axis)
#define LBLK   16          // time-block length
#define NBLK   (SEQLEN / LBLK)

typedef __attribute__((ext_vector_type(2))) float v2f;
typedef __attribute__((ext_vector_type(8))) float v8f;

__device__ __forceinline__ float csigmoid(float x) {
  // 2*sigmoid(x) - 1  (== tanh(x/2)): v_exp_f32 + v_rcp_f32 TRANS path
  return 2.0f * __builtin_amdgcn_rcpf(1.0f + __expf(-x)) - 1.0f;
}

// ---------------------------------------------------------------------------
// K0: blocked sequential phase. One workgroup = 1024 threads = 32 waves.
//
// LDS state R_s[n][a] (stride 65: conflict-free for probes (vary a) and for
// the GEMM epilogue (vary n)). Per block b (i0 = 16b):
//   probe_k = mem^k * R[a, j_k] + mem^(k-1) * sum_{u<k} g_u * G[u,k]
//   with g_u = c_u * mem^-u, G[u,k] = TM[lx_u, lx_k] (LDS-staged 16x16).
//   R update: R <- mem^16 * R + sum_u mem^(15-u) c_u TM[j_u, :]  (WMMA GEMM)
// Outputs: c_ws[t][a] and carries R_ws[b][a][n] (pre-update R).
// ---------------------------------------------------------------------------
__global__ __launch_bounds__(1024) void phase1_blocked(
    const int* __restrict__ lx, const float* __restrict__ tm,
    const float* __restrict__ diff, const float* __restrict__ eff_g,
    const float* __restrict__ mem_g, const float* __restrict__ boost_g,
    float* __restrict__ c_ws, float* __restrict__ R_ws) {
  __shared__ int   lx_s[SEQLEN + 1];
  __shared__ float dinv_s[NTASKS];
  __shared__ float Gs_s[LBLK * 17];            // G[u][k], stride 17
  __shared__ float cblk_s[LBLK * NALGOS];      // c of current block [u][a]
  __shared__ float pw_s[NALGOS * 17];          // mem_a^e, e=0..16
  __shared__ float pwi_s[NALGOS * 17];         // mem_a^-e
  __shared__ __align__(16) float tmg_s[LBLK * NTASKS];  // gathered TM rows
  __shared__ float R_s[NTASKS * 65];           // state, [n][a] stride 65

  const int tid  = threadIdx.x;
  const int wv   = tid >> 5;
  const int lane = tid & 31;
  const int hi   = lane >> 4;
  const int l15  = lane & 15;

  for (int i = tid; i < SEQLEN; i += 1024) lx_s[i] = lx[i];
  if (tid == 0) lx_s[SEQLEN] = 0;
  if (tid < NTASKS) dinv_s[tid] = 1.0f / diff[tid];
  for (int i = tid; i < NTASKS * 65; i += 1024) R_s[i] = 0.0f;

  float eff = 0.0f, boost = 0.0f;
  if (tid < NALGOS) {
    const int a = tid;
    eff   = eff_g[a];
    boost = boost_g[a];
    const float m  = mem_g[a];
    const float mi = 1.0f / m;
    float p = 1.0f, pi = 1.0f;
#pragma unroll
    for (int e = 0; e <= 16; ++e) {
      pw_s[a * 17 + e]  = p;   p  *= m;
      pwi_s[a * 17 + e] = pi;  pi *= mi;
    }
  }
  __syncthreads();

  for (int b = 0; b < NBLK; ++b) {
    // ---- stage: gathered TM rows (B operand) and 16x16 G matrix ----------
    {
      const int u  = tid >> 6;
      const int c4 = (tid & 63) * 4;
      const float4* src =
          (const float4*)(tm + (size_t)lx_s[b * LBLK + u] * NTASKS + c4);
      *(float4*)(tmg_s + u * NTASKS + c4) = *src;
    }
    if (tid < 256) {
      const int u  = tid >> 4;
      const int kk = tid & 15;
      Gs_s[u * 17 + kk] =
          tm[(size_t)lx_s[b * LBLK + u] * NTASKS + lx_s[b * LBLK + kk]];
    }
    __syncthreads();

    // ---- probe chain: 16 dependent steps, 64 threads, no barriers --------
    if (tid < NALGOS) {
      const int a = tid;
      float partial[LBLK];
#pragma unroll
      for (int k = 0; k < LBLK; ++k) partial[k] = 0.0f;
      float g[LBLK];
#pragma unroll
      for (int k = 0; k < LBLK; ++k) {
        const int jk = lx_s[b * LBLK + k];
        const float Rv = R_s[jk * 65 + a];
        float sum = partial[k];
        if (k > 0) sum += g[k - 1] * Gs_s[(k - 1) * 17 + k];
        const float probe = pw_s[a * 17 + k] * Rv +
                            (k ? pw_s[a * 17 + (k - 1)] : 0.0f) * sum;
        const float s = csigmoid(probe * dinv_s[jk]);
        const float c = eff + boost * s;
        c_ws[(b * LBLK + k) * NALGOS + a] = c;   // fire & forget
        cblk_s[k * NALGOS + a] = c;
        g[k] = c * pwi_s[a * 17 + k];
        // off-critical-path prefix updates for future steps
#pragma unroll
        for (int k2 = k + 2; k2 < LBLK; ++k2)
          partial[k2] += g[k] * Gs_s[k * 17 + k2];
      }
    }
    __syncthreads();

    // ---- block-end state update: 64x16 @ 16x256 f32 GEMM on WMMA ---------
    // A[M=a][K=u] = mem_a^(15-u) * c_u[a];  B = tmg;  64 tiles over 32 waves.
#pragma unroll
    for (int half = 0; half < 2; ++half) {
      const int Mt = (wv >> 4) + 2 * half;     // 0..3
      const int Nt = wv & 15;                  // 0..15
      const int am = Mt * 16 + l15;            // a row for A-frag lanes
      v2f af[4];
#pragma unroll
      for (int q = 0; q < 4; ++q) {
        const int u0 = q * 4 + 2 * hi;
        const int u1 = u0 + 1;
        af[q][0] = pw_s[am * 17 + (15 - u0)] * cblk_s[u0 * NALGOS + am];
        af[q][1] = pw_s[am * 17 + (15 - u1)] * cblk_s[u1 * NALGOS + am];
      }
      v8f acc = {0.f,0.f,0.f,0.f,0.f,0.f,0.f,0.f};
#pragma unroll
      for (int q = 0; q < 4; ++q) {
        const int u0 = q * 4 + 2 * hi;
        v2f bf;
        bf[0] = tmg_s[u0 * NTASKS + Nt * 16 + l15];
        bf[1] = tmg_s[(u0 + 1) * NTASKS + Nt * 16 + l15];
        acc = __builtin_amdgcn_wmma_f32_16x16x4_f32(
            false, af[q], false, bf, (short)0, acc, false, false);
      }
      const int N = Nt * 16 + l15;
#pragma unroll
      for (int r = 0; r < 8; ++r) {
        const int M = Mt * 16 + 8 * hi + r;    // C/D: M = reg + 8*(lane>=16)
        const float Rold = R_s[N * 65 + M];
        R_ws[((size_t)b * NALGOS + M) * NTASKS + N] = Rold;  // carry out
        R_s[N * 65 + M] = pw_s[M * 17 + 16] * Rold + acc[r];
      }
    }
    __syncthreads();
  }
}

// ---------------------------------------------------------------------------
// K3: WMMA output phase. res[k,n] = W @ TMg + mem^(k+1)*R_carry with
// W[k,u] = c_u * mem^(k-u) lower-triangular; 4x V_WMMA_F32_16X16X4_F32 per
// 16x16 tile. WG = 8 waves = 8 'a' values sharing the LDS-staged TM tile.
// Each lane writes 8 consecutive output floats (L2 write-combines lines).
// ---------------------------------------------------------------------------
__global__ __launch_bounds__(256) void phase2_wmma(
    const int* __restrict__ lx, const float* __restrict__ tm,
    const float* __restrict__ c_ws, const float* __restrict__ mem_g,
    const float* __restrict__ diff, const float* __restrict__ R_ws,
    float* __restrict__ out) {
  __shared__ __align__(16) float tmg[LBLK * NTASKS];  // 16 KB
  __shared__ int jrow[LBLK];

  const int bt  = blockIdx.x;       // time block 0..255
  const int ag  = blockIdx.y;       // a-group 0..7
  const int tid = threadIdx.x;

  if (tid < LBLK) jrow[tid] = lx[bt * LBLK + tid];
  __syncthreads();
  {
    const int u  = tid >> 4;
    const int c0 = (tid & 15) * 16;
    const float4* src = (const float4*)(tm + (size_t)jrow[u] * NTASKS + c0);
    float4* dst = (float4*)(tmg + u * NTASKS + c0);
    dst[0] = src[0]; dst[1] = src[1]; dst[2] = src[2]; dst[3] = src[3];
  }
  __syncthreads();

  const int wv   = tid >> 5;
  const int lane = tid & 31;
  const int a    = ag * 8 + wv;

  const float memv = mem_g[a];
  const float minv = 1.0f / memv;

  float g[LBLK];                       // g[u] = c_u * mem^-u
  {
    float mp = 1.0f;
#pragma unroll
    for (int u = 0; u < LBLK; ++u) {
      g[u] = c_ws[(bt * LBLK + u) * NALGOS + a] * mp;
      mp *= minv;
    }
  }

  const int k  = lane & 15;
  const int hi = lane >> 4;
  const float p1 = memv, p2 = p1*p1, p4 = p2*p2, p8 = p4*p4;
  float mk = 1.0f;                     // mem^k via square-multiply
  if (k & 1) mk *= p1;
  if (k & 2) mk *= p2;
  if (k & 4) mk *= p4;
  if (k & 8) mk *= p8;

  v2f afrag[4];                        // A fragments for the 4 K-chunks
#pragma unroll
  for (int q = 0; q < 4; ++q) {
    const int u0 = q * 4 + 2 * hi;     // K index of A vgpr0
    const int u1 = u0 + 1;             // K index of A vgpr1
    const float g0 = hi ? g[q*4 + 2] : g[q*4 + 0];
    const float g1 = hi ? g[q*4 + 3] : g[q*4 + 1];
    afrag[q][0] = (k >= u0) ? g0 * mk : 0.0f;
    afrag[q][1] = (k >= u1) ? g1 * mk : 0.0f;
  }

  float mpr[8];                        // mem^(r+1)
  mpr[0] = memv;
#pragma unroll
  for (int r = 1; r < 8; ++r) mpr[r] = mpr[r-1] * memv;
  const float phi = hi ? p8 : 1.0f;    // extra mem^8 for upper output rows
  const int nlo = lane & 15;

  for (int nt = 0; nt < 16; ++nt) {
    const int n = nt * 16 + nlo;
    v8f acc = {0.f,0.f,0.f,0.f,0.f,0.f,0.f,0.f};
#pragma unroll
    for (int q = 0; q < 4; ++q) {
      const int u0 = q * 4 + 2 * hi;
      v2f bf;
      bf[0] = tmg[u0 * NTASKS + n];
      bf[1] = tmg[(u0 + 1) * NTASKS + n];
      acc = __builtin_amdgcn_wmma_f32_16x16x4_f32(
          false, afrag[q], false, bf, (short)0, acc, false, false);
    }
    const float Rc = R_ws[((size_t)bt * NALGOS + a) * NTASKS + n] * phi;
    const float dv = __builtin_amdgcn_rcpf(diff[n]);
    float* op = out + (size_t)(a * NTASKS + n) * TP1 + bt * LBLK + 1 + 8 * hi;
#pragma unroll
    for (int r = 0; r < 8; ++r) {
      const float resv = acc[r] + mpr[r] * Rc;   // M = r + 8*hi
      op[r] = csigmoid(resv * dv);               // 8 consecutive floats
    }
  }
}

// ---------------------------------------------------------------------------
// K4: out[a, n, 0] = 0
// ---------------------------------------------------------------------------
__global__ __launch_bounds__(256) void zero_first(float* __restrict__ out) {
  const size_t idx = (size_t)blockIdx.x * 256 + threadIdx.x;
  out[idx * TP1] = 0.0f;
}

// ---------------------------------------------------------------------------
extern "C" void kernel_launch(void* const* d_in, const int* in_sizes, int n_in,
                              void* d_out, int out_size, void* d_ws, size_t ws_size,
                              hipStream_t stream) {
  const int*   lx    = (const int*)d_in[0];
  const float* tm    = (const float*)d_in[1];
  const float* diff  = (const float*)d_in[2];
  const float* eff   = (const float*)d_in[3];
  const float* memv  = (const float*)d_in[4];
  const float* boost = (const float*)d_in[5];
  float* out = (float*)d_out;

  // workspace: c (1 MB) | R carries (16 MB)
  float* c_ws = (float*)d_ws;
  float* R_ws = c_ws + (size_t)SEQLEN * NALGOS;

  phase1_blocked<<<1, 1024, 0, stream>>>(lx, tm, diff, eff, memv, boost,
                                         c_ws, R_ws);
  zero_first<<<NALGOS, NTASKS, 0, stream>>>(out);
  phase2_wmma<<<dim3(NBLK, 8), 256, 0, stream>>>(lx, tm, c_ws, memv, diff,
                                                 R_ws, out);
}